// MultiCueCoarseGate_14998025798251
// MI455X (gfx1250) — compile-verified
//
#include <hip/hip_runtime.h>
#include <hip/hip_bf16.h>
#include <math.h>
#include <stdint.h>

// ---------------- MultiCueCoarseGate: streaming per-patch cues ----------------
// One workgroup per 256x256 patch. 32-row bands double-buffered in LDS via
// CDNA5 async global->LDS DMA (global_load_async_to_lds_b128 / s_wait_asynccnt).

#define ROWSTRIDE 264          // floats per LDS row (data at [4..259], pads at [3],[260])
#define PADL 4                 // data start offset within a row (keeps 16B alignment)
#define BANDROWS 34            // 32 interior + 2 halo rows
#define BUFFLOATS (BANDROWS * ROWSTRIDE)

__device__ __forceinline__ void async_ld16(unsigned lds_off, const float* gptr) {
    // LDS-dest async load: vdst = 32-bit LDS byte offset, vaddr = 64-bit global addr
    asm volatile("global_load_async_to_lds_b128 %0, %1, off"
                 :: "v"(lds_off), "v"(gptr)
                 : "memory");
}

__device__ __forceinline__ void wait_async0() {
    asm volatile("s_wait_asynccnt 0" ::: "memory");
}

__device__ __forceinline__ unsigned lds_off32(const float* p) {
    // low 32 bits of a generic pointer to __shared__ == LDS byte offset
    return (unsigned)(uintptr_t)p;
}

__device__ __forceinline__ void load_band(const float* __restrict__ src, int R,
                                          float* buf, int tid) {
    // interior rows R..R+31 -> lds rows 1..32 : 32*64 = 2048 float4 chunks, 8/thread
#pragma unroll
    for (int it = 0; it < 8; ++it) {
        int idx  = it * 256 + tid;
        int lrow = idx >> 6;        // 0..31
        int c4   = idx & 63;
        float* dst = buf + (lrow + 1) * ROWSTRIDE + PADL + c4 * 4;
        async_ld16(lds_off32(dst), src + (size_t)(R + lrow) * 256 + c4 * 4);
    }
    // halo rows R-1 (lds row 0) and R+32 (lds row 33): 128 chunks, threads 0..127
    if (tid < 128) {
        int h    = tid >> 6;                 // 0 = top halo, 1 = bottom halo
        int c4   = tid & 63;
        int grow = h ? (R + 32) : (R - 1);
        int lrow = h ? (BANDROWS - 1) : 0;
        float* dst = buf + lrow * ROWSTRIDE + PADL + c4 * 4;
        if (grow >= 0 && grow < 256) {
            async_ld16(lds_off32(dst), src + (size_t)grow * 256 + c4 * 4);
        } else {
            dst[0] = 0.0f; dst[1] = 0.0f; dst[2] = 0.0f; dst[3] = 0.0f;  // zero pad row
        }
    }
}

__global__ void __launch_bounds__(256)
multi_cue_gate_kernel(const float* __restrict__ patches, float* __restrict__ out, int B) {
    const int b   = blockIdx.x;
    const int tid = threadIdx.x;
    const float* src = patches + (size_t)b * 65536;

    __shared__ __align__(16) float    tile[2 * BUFFLOATS];
    __shared__ unsigned               hist[256];
    __shared__ double                 redd[256];
    __shared__ float                  redf[256];

    // init: histogram zeros + zero the column pads of both buffers (never overwritten)
    hist[tid] = 0u;
    if (tid < 2 * BANDROWS * 2) {           // 136 pad slots
        int p    = tid;
        int bufi = (p >= 2 * BANDROWS) ? 1 : 0;
        p       -= bufi * 2 * BANDROWS;
        int row  = p >> 1;
        int side = p & 1;
        tile[bufi * BUFFLOATS + row * ROWSTRIDE + (side ? (PADL + 256) : (PADL - 1))] = 0.0f;
    }

    // prologue: DMA band 0 into buffer 0
    load_band(src, 0, tile, tid);
    wait_async0();
    __syncthreads();

    double sum = 0.0, sumsq = 0.0;
    float  emax = 0.0f;
    const int c  = tid;                      // this thread owns column c
    const int o0 = PADL - 1 + c, o1 = PADL + c, o2 = PADL + 1 + c;

    for (int band = 0; band < 8; ++band) {
        const int bi = band & 1;
        if (band + 1 < 8)
            load_band(src, (band + 1) * 32, tile + ((band + 1) & 1) * BUFFLOATS, tid);

        const float* base = tile + bi * BUFFLOATS;
        // 3x3 sliding window down the band for column c
        float t0 = base[o0],             t1 = base[o1],             t2 = base[o2];
        float m0 = base[ROWSTRIDE + o0], m1 = base[ROWSTRIDE + o1], m2 = base[ROWSTRIDE + o2];
#pragma unroll 4
        for (int k = 1; k <= 32; ++k) {
            const float* br = base + (k + 1) * ROWSTRIDE;
            float b0 = br[o0], b1 = br[o1], b2 = br[o2];

            // cue 1 partials on the window center m1 (each pixel counted once)
            double x = (double)m1;
            sum   += x;
            sumsq += x * x;

            // cue 2: u8-quantized histogram (truncation matches jnp int32 cast)
            float q  = fminf(fmaxf(m1, 0.0f), 1.0f) * 255.0f;
            int  bin = (int)q; bin = bin > 255 ? 255 : (bin < 0 ? 0 : bin);
            atomicAdd(&hist[bin], 1u);

            // cue 3: Sobel cross-correlation (XLA conv does not flip kernels)
            float gx = (t2 - t0) + 2.0f * (m2 - m0) + (b2 - b0);
            float gy = (b0 + 2.0f * b1 + b2) - (t0 + 2.0f * t1 + t2);
            emax = fmaxf(emax, fmaxf(fabsf(gx), fabsf(gy)));

            t0 = m0; t1 = m1; t2 = m2;
            m0 = b0; m1 = b1; m2 = b2;
        }
        wait_async0();       // band+1's DMA complete (this wave's issues, in order)
        __syncthreads();     // all waves done computing band & all DMA visible
    }

    // ---- workgroup reductions ----
    redd[tid] = sum; __syncthreads();
    for (int s = 128; s > 0; s >>= 1) { if (tid < s) redd[tid] += redd[tid + s]; __syncthreads(); }
    double S1 = redd[0]; __syncthreads();

    redd[tid] = sumsq; __syncthreads();
    for (int s = 128; s > 0; s >>= 1) { if (tid < s) redd[tid] += redd[tid + s]; __syncthreads(); }
    double S2 = redd[0]; __syncthreads();

    redf[tid] = emax; __syncthreads();
    for (int s = 128; s > 0; s >>= 1) { if (tid < s) redf[tid] = fmaxf(redf[tid], redf[tid + s]); __syncthreads(); }
    float E = redf[0]; __syncthreads();

    // entropy: total count is exactly 65536, matching reference's max(sum, EPS)
    unsigned h = hist[tid];
    float p    = (float)h * (1.0f / 65536.0f);
    float term = (h > 0u) ? (-p * log2f(p)) : 0.0f;
    redf[tid] = term; __syncthreads();
    for (int s = 128; s > 0; s >>= 1) { if (tid < s) redf[tid] += redf[tid + s]; __syncthreads(); }
    float H = redf[0];

    if (tid == 0) {
        const double N = 65536.0;
        float var = (float)((S2 - S1 * S1 / N) / (N - 1.0));   // ddof=1
        out[0 * B + b] = var;
        out[1 * B + b] = H;
        out[2 * B + b] = E;
        out[3 * B + b] = 0.0f;                                  // mahalanobis placeholder
    }
}

extern "C" void kernel_launch(void* const* d_in, const int* in_sizes, int n_in,
                              void* d_out, int out_size, void* d_ws, size_t ws_size,
                              hipStream_t stream) {
    const float* patches = (const float*)d_in[0];
    float* out = (float*)d_out;
    const int B = in_sizes[0] >> 16;     // elements / 65536 patches
    multi_cue_gate_kernel<<<dim3(B), dim3(256), 0, stream>>>(patches, out, B);
}